// Mosaic2DBase_53523882443321
// MI455X (gfx1250) — compile-verified
//
#include <hip/hip_runtime.h>
#include <stdint.h>

// Mosaic2D pick + mask tiling for MI455X (gfx1250).
// out[b,h,w] = x[b, 4*(h%4)+(w%4), h, w] * msfa[c, h%4, w%4]   (msfa is one-hot eye(16))
// w  [c,h,w] = msfa[c, h%4, w%4]
//
// Memory-bound: ~224 MB total traffic -> ~10us at 23.3 TB/s. Strategy:
//  - async-copy the 4 needed channel rows per output row into LDS (contiguous
//    b128 per lane -> every fetched DRAM line fully consumed = traffic floor),
//  - do the stride-16B pick out of LDS (banked, conflict-cheap),
//  - non-temporal b128 stores for both outputs (never re-read).

#define HH 1024
#define WW 1024
#define CH 16
#define BATCH 8

// native clang vector type so __builtin_nontemporal_store lowers to one b128 store
typedef float v4f __attribute__((ext_vector_type(4)));

// ---------------- Kernel 1: w = tile(msfa) over [16,1024,1024] ----------------
__global__ __launch_bounds__(256) void msfa_tile_w(const float* __restrict__ msfa,
                                                   float* __restrict__ wout) {
  uint32_t idx = blockIdx.x * 256u + threadIdx.x;   // over 16*1024*256 float4 groups
  uint32_t h = (idx >> 8) & (HH - 1);
  uint32_t c = idx >> 18;
  // msfa[c][h%4][0..3] is a contiguous 16-byte group
  const v4f* m4 = (const v4f*)msfa;
  v4f v = m4[c * 4u + (h & 3u)];
  __builtin_nontemporal_store(v, ((v4f*)wout) + idx);
}

// ---------------- Kernel 2: mosaic pick via async LDS staging ----------------
__global__ __launch_bounds__(256) void msfa_pick(const float* __restrict__ x,
                                                 const float* __restrict__ msfa,
                                                 float* __restrict__ out) {
  __shared__ float smem[4 * WW];                    // 4 channel rows, 16 KB

  const uint32_t t  = threadIdx.x;                  // 0..255
  const uint32_t bh = blockIdx.x;                   // b*1024 + h
  const uint32_t h  = bh & (HH - 1);
  const uint32_t b  = bh >> 10;
  const uint32_t r  = h & 3u;                       // tile row -> channels 4r..4r+3

  // LDS byte address of smem (low 32 bits of the generic pointer = LDS offset)
  const uint32_t lds_base = (uint32_t)(uintptr_t)&smem[0];
  const uint32_t voff     = t << 4;                 // 16 bytes per lane, contiguous

  // base of channel (4r+0) row [b, 4r, h, 0]
  const float* row0 = x + (((size_t)(b * CH + 4u * r) * HH + h) * WW);

  // Async-stage 4 channel rows (4 KB each) into LDS. GVS mode:
  //   mem = SADDR(64) + VADDR(32), LDS dst = VDST vgpr. Tracked by ASYNCcnt.
#pragma unroll
  for (int j = 0; j < 4; ++j) {
    uint32_t la = lds_base + (uint32_t)(j * (WW * 4)) + voff;
    uint64_t sa = (uint64_t)(uintptr_t)(row0 + (size_t)j * HH * WW);
    asm volatile("global_load_async_to_lds_b128 %0, %1, %2"
                 :
                 : "v"(la), "v"(voff), "s"(sa)
                 : "memory");
  }

  // Overlap: fetch the 4 diagonal mask scales (==1.0 for eye) while DMA runs.
  float s0 = msfa[(4u * r + 0u) * 16u + r * 4u + 0u];
  float s1 = msfa[(4u * r + 1u) * 16u + r * 4u + 1u];
  float s2 = msfa[(4u * r + 2u) * 16u + r * 4u + 2u];
  float s3 = msfa[(4u * r + 3u) * 16u + r * 4u + 3u];

  asm volatile("s_wait_asynccnt 0x0" ::: "memory");
  __syncthreads();

  // Pick: out[p+j] = row(4r+j)[p+j],  p = 4t.  Stride-16B reads out of LDS.
  const uint32_t p = t << 2;
  v4f o;
  o.x = smem[0 * WW + p + 0] * s0;
  o.y = smem[1 * WW + p + 1] * s1;
  o.z = smem[2 * WW + p + 2] * s2;
  o.w = smem[3 * WW + p + 3] * s3;

  v4f* orow = (v4f*)(out + (((size_t)b * HH + h) * WW));
  __builtin_nontemporal_store(o, orow + t);
}

extern "C" void kernel_launch(void* const* d_in, const int* in_sizes, int n_in,
                              void* d_out, int out_size, void* d_ws, size_t ws_size,
                              hipStream_t stream) {
  const float* x    = (const float*)d_in[0];   // [8,16,1024,1024] f32
  const float* msfa = (const float*)d_in[1];   // [16,4,4] f32
  float* out  = (float*)d_out;                          // [8,1024,1024]
  float* wout = out + (size_t)BATCH * HH * WW;          // [16,1024,1024]

  msfa_pick<<<BATCH * HH, 256, 0, stream>>>(x, msfa, out);

  const uint32_t w_groups = (CH * HH * (WW / 4)) / 256; // 16384 blocks
  msfa_tile_w<<<w_groups, 256, 0, stream>>>(msfa, wout);
}